// ReconLoss_58119497449688
// MI455X (gfx1250) — compile-verified
//
#include <hip/hip_runtime.h>

// ============================================================================
// ReconLoss on gfx1250 (MI455X): fused SIM-reconstruction loss.
//   sup = illum^2 ; psf = |2 J1(sR)/(sR)|^2 ; M = |FFT2(psf)|/sum(psf)
//   ST  = IFFT2( FFT2(sup) * M ) / (H*W) ; out = mean|inputs - (ST + noisy)|
// 1-D FFT(1024) = four-step 32x32: Z = F32*X ; Z *= W1024^(k1*n2) ; Y = Z*F32.
// Each 32x32 complex matmul runs on one wave32 via v_wmma_f32_16x16x4_f32.
// pred_img (d_in[1]) is mathematically unused (torch aliasing in reference).
// Workspace use: 6 planes of 32MB + tables/partials  (~202 MB).
// ============================================================================

typedef float v2f __attribute__((ext_vector_type(2)));
typedef float v8f __attribute__((ext_vector_type(8)));

namespace {
constexpr int   kB    = 8;
constexpr int   kN    = 1024;
constexpr int   kNPix = kB * kN * kN;   // 8,388,608
constexpr int   kP    = 33;             // padded LDS pitch (33 odd -> few bank conflicts)
constexpr float kPI   = 3.14159265358979323846f;
constexpr float kEPS32 = 1.1920928955078125e-07f;
}

// ---------------------------------------------------------------------------
// Numerical-Recipes J1 approximation (float), matches reference math.
// ---------------------------------------------------------------------------
__device__ __forceinline__ float bessel_j1f(float x) {
  float ax = fabsf(x);
  float y  = x * x;
  float num = x * (72362614232.0f + y * (-7895059235.0f + y * (242396853.1f +
              y * (-2972611.439f + y * (15704.48260f + y * (-30.16036606f))))));
  float den = 144725228442.0f + y * (2300535178.0f + y * (18583304.74f +
              y * (99447.43394f + y * (376.9991397f + y))));
  float small = num / den;
  float axs = fmaxf(ax, 1e-20f);
  float z   = 8.0f / axs;
  float y2  = z * z;
  float xx  = axs - 2.356194491f;
  float p1 = 1.0f + y2 * (0.183105e-2f + y2 * (-0.3516396496e-4f +
             y2 * (0.2457520174e-5f + y2 * (-0.240337019e-6f))));
  float p2 = 0.04687499995f + y2 * (-0.2002690873e-3f + y2 * (0.8449199096e-5f +
             y2 * (-0.88228987e-6f + y2 * 0.105787412e-6f)));
  float sgn = (x < 0.0f) ? -1.0f : 1.0f;
  float big = sqrtf(0.636619772f / axs) * (cosf(xx) * p1 - z * sinf(xx) * p2) * sgn;
  return (ax < 8.0f) ? small : big;
}

// ---------------------------------------------------------------------------
// One-wave 32x32 complex matmul C = A * B on LDS (pitch kP) via WMMA f32.
// Fragment layouts per CDNA5 ISA (16x4 f32 A-matrix; 16x16 f32 C/D).
//   Re(C) += Ar*Br + (-Ai)*Bi ;  Im(C) += Ar*Bi + Ai*Br
// 128 v_wmma_f32_16x16x4_f32 per call.
// ---------------------------------------------------------------------------
__device__ __forceinline__ void cmatmul32(const float* __restrict__ Ar,
                                          const float* __restrict__ Ai,
                                          const float* __restrict__ Br,
                                          const float* __restrict__ Bi,
                                          float* __restrict__ Cr,
                                          float* __restrict__ Ci,
                                          int lane) {
  const int ml = lane & 15;   // row (A/D) or col (B/D) within tile
  const int lg = lane >> 4;   // half-wave group
  for (int mt = 0; mt < 2; ++mt) {
    for (int nt = 0; nt < 2; ++nt) {
      v8f dr = {};
      v8f di = {};
      const int m0 = mt * 16, n0 = nt * 16;
#pragma unroll
      for (int kc = 0; kc < 8; ++kc) {
        const int ka = kc * 4 + lg * 2;   // K pair owned by this half-wave
        v2f arf, aif, anf, brf, bif;
        arf[0] = Ar[(m0 + ml) * kP + ka];
        arf[1] = Ar[(m0 + ml) * kP + ka + 1];
        aif[0] = Ai[(m0 + ml) * kP + ka];
        aif[1] = Ai[(m0 + ml) * kP + ka + 1];
        anf[0] = -aif[0];
        anf[1] = -aif[1];
        brf[0] = Br[(ka    ) * kP + n0 + ml];
        brf[1] = Br[(ka + 1) * kP + n0 + ml];
        bif[0] = Bi[(ka    ) * kP + n0 + ml];
        bif[1] = Bi[(ka + 1) * kP + n0 + ml];
        dr = __builtin_amdgcn_wmma_f32_16x16x4_f32(false, arf, false, brf, (short)0, dr, false, false);
        dr = __builtin_amdgcn_wmma_f32_16x16x4_f32(false, anf, false, bif, (short)0, dr, false, false);
        di = __builtin_amdgcn_wmma_f32_16x16x4_f32(false, arf, false, bif, (short)0, di, false, false);
        di = __builtin_amdgcn_wmma_f32_16x16x4_f32(false, aif, false, brf, (short)0, di, false, false);
      }
#pragma unroll
      for (int j = 0; j < 8; ++j) {   // D: VGPR j -> rows j / j+8
        const int r = m0 + j + lg * 8;
        const int c = n0 + ml;
        Cr[r * kP + c] = dr[j];
        Ci[r * kP + c] = di[j];
      }
    }
  }
}

// ---------------------------------------------------------------------------
// Tables: F32[k][n] = exp(-2pi i k n / 32), Tw[k1][n2] = exp(-2pi i k1 n2 / 1024)
// Layout in ws: [Fre(1024) | Fim(1024) | Twre(1024) | Twim(1024)]
// ---------------------------------------------------------------------------
__global__ __launch_bounds__(256) void k_tables(float* __restrict__ tab) {
  for (int i = threadIdx.x; i < 1024; i += 256) {
    const int r = i >> 5, c = i & 31;
    const float a32 = -2.0f * kPI * (float)(r * c) / 32.0f;
    const float a1k = -2.0f * kPI * (float)(r * c) / 1024.0f;
    tab[i]        = cosf(a32);
    tab[1024 + i] = sinf(a32);
    tab[2048 + i] = cosf(a1k);
    tab[3072 + i] = sinf(a1k);
  }
}

// ---------------------------------------------------------------------------
// Generate sup = illum^2 and psf; per-block psf partial sums (deterministic).
// ---------------------------------------------------------------------------
__global__ __launch_bounds__(256) void k_gen(const float* __restrict__ meanInt,
                                             const float* __restrict__ wavev,
                                             const float* __restrict__ phsh,
                                             const float* __restrict__ modf,
                                             const float* __restrict__ psfScale,
                                             float* __restrict__ supRe, float* __restrict__ supIm,
                                             float* __restrict__ psfRe, float* __restrict__ psfIm,
                                             float* __restrict__ partial) {
  const int idx = blockIdx.x * 256 + threadIdx.x;
  const int b   = idx >> 20;
  const int pix = idx & ((1 << 20) - 1);
  const int r   = pix >> 10;      // Xg for illum, Yg for psf (indexing='ij')
  const int c   = pix & 1023;
  const int Na = 3, Ns = 3, ia = 2, isf = 2;  // only last angle/shift survives

  const float mi  = meanInt[b * Na + ia];
  const float mf  = modf[b * Na + ia];
  const float wv0 = wavev[(b * Na + ia) * 2 + 0];
  const float wv1 = wavev[(b * Na + ia) * 2 + 1];
  const float ps  = phsh[(b * Na + ia) * Ns + isf];
  const float wo  = 512.0f;
  const float phase = 2.0f * kPI * (wv0 * ((float)r - wo) + wv1 * ((float)c - wo)) + ps;
  const float cv = fminf(fmaxf(cosf(phase), 0.0f), 1.0f);
  const float il = mi + mi * mf * cv;
  supRe[idx] = il * il;
  supIm[idx] = 0.0f;

  const float R   = sqrtf((float)(r * r + c * c));
  const float arg = psfScale[b] * R + kEPS32;
  const float t   = 2.0f * bessel_j1f(arg) / arg;
  const float p   = t * t;
  psfRe[idx] = p;
  psfIm[idx] = 0.0f;

  __shared__ float sred[256];
  sred[threadIdx.x] = p;
  __syncthreads();
  for (int k = 128; k > 0; k >>= 1) {
    if (threadIdx.x < k) sred[threadIdx.x] += sred[threadIdx.x + k];
    __syncthreads();
  }
  if (threadIdx.x == 0) partial[blockIdx.x] = sred[0];  // 4096 blocks per batch
}

__global__ __launch_bounds__(256) void k_psfsum(const float* __restrict__ partial,
                                                float* __restrict__ psfSum) {
  __shared__ float s[256];
  const int b = blockIdx.x;
  float acc = 0.0f;
  for (int i = threadIdx.x; i < 4096; i += 256) acc += partial[b * 4096 + i];
  s[threadIdx.x] = acc;
  __syncthreads();
  for (int k = 128; k > 0; k >>= 1) {
    if (threadIdx.x < k) s[threadIdx.x] += s[threadIdx.x + k];
    __syncthreads();
  }
  if (threadIdx.x == 0) psfSum[b] = s[0];
}

// ---------------------------------------------------------------------------
// In-place length-1024 FFT along contiguous rows (B*H rows). One wave per row,
// 2 waves per block. Four-step: Z = F*X ; twiddle ; Y = Z*F ; digit-swap out.
// inverse: conjugated tables + 1/1024 scale (two passes -> 1/(H*W)).
// ---------------------------------------------------------------------------
__global__ __launch_bounds__(64) void k_fft1024_rows(float* __restrict__ re,
                                                     float* __restrict__ im,
                                                     const float* __restrict__ tab,
                                                     int inverse) {
  __shared__ float sF[2][32 * kP];        // DFT32 (block-shared)
  __shared__ float sT[2][32 * kP];        // twiddle (block-shared)
  __shared__ float sA[2][2][32 * kP];     // per-wave scratch X / Y
  __shared__ float sB[2][2][32 * kP];     // per-wave scratch Z

  const int tid  = threadIdx.x;
  const int wv   = tid >> 5;
  const int lane = tid & 31;
  const float cs = inverse ? -1.0f : 1.0f;   // conjugate tables for inverse

  for (int i = tid; i < 1024; i += 64) {
    const int r = i >> 5, c = i & 31;
    sF[0][r * kP + c] = tab[i];
    sF[1][r * kP + c] = cs * tab[1024 + i];
    sT[0][r * kP + c] = tab[2048 + i];
    sT[1][r * kP + c] = cs * tab[3072 + i];
  }
  __syncthreads();

  float* ar = sA[wv][0];
  float* ai = sA[wv][1];
  float* br = sB[wv][0];
  float* bi = sB[wv][1];

  const size_t row  = (size_t)blockIdx.x * 2 + wv;
  const size_t base = row * 1024;

  // X[n1][n2] = x[32*n1 + n2]
  for (int i = lane; i < 1024; i += 32) {
    const int n1 = i >> 5, n2 = i & 31;
    ar[n1 * kP + n2] = re[base + i];
    ai[n1 * kP + n2] = im[base + i];
  }

  cmatmul32(sF[0], sF[1], ar, ai, br, bi, lane);   // Z = F * X

  for (int i = lane; i < 1024; i += 32) {          // Z *= W1024^(k1*n2)
    const int id = (i >> 5) * kP + (i & 31);
    const float twr = sT[0][id], twi = sT[1][id];
    const float zr = br[id], zi = bi[id];
    br[id] = zr * twr - zi * twi;
    bi[id] = zr * twi + zi * twr;
  }

  cmatmul32(br, bi, sF[0], sF[1], ar, ai, lane);   // Y = Z' * F

  const float scl = inverse ? (1.0f / 1024.0f) : 1.0f;
  for (int i = lane; i < 1024; i += 32) {          // y[k1 + 32*k2] = Y[k1][k2]
    const int k1 = i & 31, k2 = i >> 5;
    const int id = k1 * kP + k2;
    re[base + i] = ar[id] * scl;
    im[base + i] = ai[id] * scl;
  }
}

// ---------------------------------------------------------------------------
// Batched coalesced complex transpose (per-batch 1024x1024), 32x32 LDS tiles.
// ---------------------------------------------------------------------------
__global__ __launch_bounds__(256) void k_transpose(const float* __restrict__ sre,
                                                   const float* __restrict__ sim,
                                                   float* __restrict__ dre,
                                                   float* __restrict__ dimg) {
  __shared__ float tr[32][kP];
  __shared__ float ti[32][kP];
  const size_t off = (size_t)blockIdx.z * kN * kN;
  int x = blockIdx.x * 32 + threadIdx.x;
  int y = blockIdx.y * 32 + threadIdx.y;
  for (int j = 0; j < 32; j += 8) {
    tr[threadIdx.y + j][threadIdx.x] = sre[off + (size_t)(y + j) * kN + x];
    ti[threadIdx.y + j][threadIdx.x] = sim[off + (size_t)(y + j) * kN + x];
  }
  __syncthreads();
  x = blockIdx.y * 32 + threadIdx.x;
  y = blockIdx.x * 32 + threadIdx.y;
  for (int j = 0; j < 32; j += 8) {
    dre[off + (size_t)(y + j) * kN + x]  = tr[threadIdx.x][threadIdx.y + j];
    dimg[off + (size_t)(y + j) * kN + x] = ti[threadIdx.x][threadIdx.y + j];
  }
}

// M = |FFT2(psf)| / sum(psf)   (max|OTF| == DC == sum since psf >= 0)
__global__ __launch_bounds__(256) void k_mag(const float* __restrict__ tre,
                                             const float* __restrict__ tim,
                                             const float* __restrict__ psfSum,
                                             float* __restrict__ M) {
  const int idx = blockIdx.x * 256 + threadIdx.x;
  const float s = psfSum[idx >> 20];
  const float vr = tre[idx], vi = tim[idx];
  M[idx] = sqrtf(vr * vr + vi * vi) / s;
}

__global__ __launch_bounds__(256) void k_apply(const float* __restrict__ M,
                                               float* __restrict__ re,
                                               float* __restrict__ im) {
  const int idx = blockIdx.x * 256 + threadIdx.x;
  const float m = M[idx];
  re[idx] *= m;
  im[idx] *= m;
}

// |inputs - (Re(ST) + noisy)| -> per-block partials (deterministic).
__global__ __launch_bounds__(256) void k_absdiff(const float* __restrict__ inp,
                                                 const float* __restrict__ noisy,
                                                 const float* __restrict__ stRe,
                                                 float* __restrict__ partial) {
  __shared__ float s[256];
  const int idx = blockIdx.x * 256 + threadIdx.x;
  __builtin_prefetch(inp + idx + 8192, 0, 0);  // global_prefetch_b8 (speculative)
  s[threadIdx.x] = fabsf(inp[idx] - (stRe[idx] + noisy[idx]));
  __syncthreads();
  for (int k = 128; k > 0; k >>= 1) {
    if (threadIdx.x < k) s[threadIdx.x] += s[threadIdx.x + k];
    __syncthreads();
  }
  if (threadIdx.x == 0) partial[blockIdx.x] = s[0];
}

__global__ __launch_bounds__(256) void k_final(const float* __restrict__ partial,
                                               float* __restrict__ out) {
  __shared__ float s[256];
  float acc = 0.0f;
  for (int i = threadIdx.x; i < 32768; i += 256) acc += partial[i];
  s[threadIdx.x] = acc;
  __syncthreads();
  for (int k = 128; k > 0; k >>= 1) {
    if (threadIdx.x < k) s[threadIdx.x] += s[threadIdx.x + k];
    __syncthreads();
  }
  if (threadIdx.x == 0) out[0] = s[0] * (1.0f / (float)kNPix);
}

// ===========================================================================
extern "C" void kernel_launch(void* const* d_in, const int* in_sizes, int n_in,
                              void* d_out, int out_size, void* d_ws, size_t ws_size,
                              hipStream_t stream) {
  (void)in_sizes; (void)n_in; (void)out_size; (void)ws_size;
  const float* inputs   = (const float*)d_in[0];
  // d_in[1] = pred_img: mathematically unused (see header comment)
  const float* noisy    = (const float*)d_in[2];
  const float* meanInt  = (const float*)d_in[3];
  const float* wavev    = (const float*)d_in[4];
  const float* phsh     = (const float*)d_in[5];
  const float* modf     = (const float*)d_in[6];
  const float* psfScale = (const float*)d_in[7];
  float* out = (float*)d_out;

  float* w = (float*)d_ws;
  const size_t NP = (size_t)kNPix;
  float* supRe = w;
  float* supIm = w + NP;
  float* psfRe = w + 2 * NP;
  float* psfIm = w + 3 * NP;
  float* tmpRe = w + 4 * NP;
  float* tmpIm = w + 5 * NP;
  float* tab     = w + 6 * NP;       // 4096 floats
  float* psfSum  = tab + 4096;       // 8 floats
  float* partial = psfSum + 8;       // 32768 floats

  const int nElemBlocks = kNPix / 256;    // 32768
  const int nFftBlocks  = (kB * kN) / 2;  // 4096 (2 rows per block)
  dim3 tBlock(32, 8);
  dim3 tGrid(32, 32, kB);

  k_tables<<<dim3(1), dim3(256), 0, stream>>>(tab);
  k_gen<<<dim3(nElemBlocks), dim3(256), 0, stream>>>(meanInt, wavev, phsh, modf, psfScale,
                                                     supRe, supIm, psfRe, psfIm, partial);
  k_psfsum<<<dim3(kB), dim3(256), 0, stream>>>(partial, psfSum);

  // FFT2(psf) (ends in transposed layout) -> M into psfRe
  k_fft1024_rows<<<dim3(nFftBlocks), dim3(64), 0, stream>>>(psfRe, psfIm, tab, 0);
  k_transpose<<<tGrid, tBlock, 0, stream>>>(psfRe, psfIm, tmpRe, tmpIm);
  k_fft1024_rows<<<dim3(nFftBlocks), dim3(64), 0, stream>>>(tmpRe, tmpIm, tab, 0);
  k_mag<<<dim3(nElemBlocks), dim3(256), 0, stream>>>(tmpRe, tmpIm, psfSum, psfRe);

  // FFT2(sup) -> tmp (transposed layout, same as M)
  k_fft1024_rows<<<dim3(nFftBlocks), dim3(64), 0, stream>>>(supRe, supIm, tab, 0);
  k_transpose<<<tGrid, tBlock, 0, stream>>>(supRe, supIm, tmpRe, tmpIm);
  k_fft1024_rows<<<dim3(nFftBlocks), dim3(64), 0, stream>>>(tmpRe, tmpIm, tab, 0);

  k_apply<<<dim3(nElemBlocks), dim3(256), 0, stream>>>(psfRe, tmpRe, tmpIm);

  // IFFT2 -> sup in standard [b][y][x] layout; Re(sup) = ST
  k_fft1024_rows<<<dim3(nFftBlocks), dim3(64), 0, stream>>>(tmpRe, tmpIm, tab, 1);
  k_transpose<<<tGrid, tBlock, 0, stream>>>(tmpRe, tmpIm, supRe, supIm);
  k_fft1024_rows<<<dim3(nFftBlocks), dim3(64), 0, stream>>>(supRe, supIm, tab, 1);

  k_absdiff<<<dim3(nElemBlocks), dim3(256), 0, stream>>>(inputs, noisy, supRe, partial);
  k_final<<<dim3(1), dim3(256), 0, stream>>>(partial, out);
}